// MeanPooling_33371895890588
// MI455X (gfx1250) — compile-verified
//
#include <hip/hip_runtime.h>

// MeanPooling (segment mean, sorted int index) for MI455X / gfx1250.
//
// Roofline: 268 MB streamed (x fp32 + idx i32), ~1.2 MB resident accumulator
// traffic -> ~11.5 us floor at 23.3 TB/s. Pure HBM-bandwidth bound; WMMA does
// not apply (fp32 irregular segmented reduction). CDNA5 features used:
//  - global_load_async_to_lds_b128 (ASYNCcnt data mover) double-buffered
//    streaming, per-wave-private LDS windows (no barriers), wave32 tiling
//  - th:TH_LOAD_NT on the stream so the 800 KB L2-resident segment
//    accumulator (hit by ~1.25M no-return global_atomic_add_f32) is not
//    evicted by the 268 MB one-pass stream (> 192 MB L2)

#define NSEG   100000
#define NTOT   33554432
#define TPB    256
#define EPT    64            // contiguous elements owned per thread
#define STAGES (EPT / 4)     // 4 elements (16B x + 16B idx) per lane per stage

__device__ __forceinline__ unsigned lds_off(const void* p) {
  // addrspace(3) -> generic puts the LDS byte offset in the low 32 bits
  // (shared aperture occupies [63:32]); truncation yields the DS offset.
  return (unsigned)(unsigned long long)p;
}

__device__ __forceinline__ void async_cp16_nt(unsigned lds_byte, const void* gaddr) {
  // CDNA5 async data mover: 16B per lane, global -> LDS, tracked by ASYNCcnt.
  // Non-temporal: one-pass stream, keep L2 for the atomic accumulator.
  asm volatile("global_load_async_to_lds_b128 %0, %1, off th:TH_LOAD_NT"
               :: "v"(lds_byte), "v"((unsigned long long)gaddr)
               : "memory");
}

__global__ __launch_bounds__(TPB) void zero_ws_kernel(float* __restrict__ ws, int n) {
  int i = blockIdx.x * blockDim.x + threadIdx.x;
  if (i < n) ws[i] = 0.0f;
}

__global__ __launch_bounds__(TPB) void segsum_kernel(const float* __restrict__ x,
                                                     const int*   __restrict__ idx,
                                                     float* __restrict__ seg_sum,
                                                     float* __restrict__ seg_cnt) {
  // Per-wave-private double buffer: [x: 512B][idx: 512B] x 2 = 2KB per wave.
  __shared__ __align__(16) unsigned char lds[(TPB / 32) * 2048];

  const int lane = threadIdx.x & 31;
  const int wave = threadIdx.x >> 5;
  unsigned char* wbase = lds + wave * 2048;

  const long long tid  = (long long)blockIdx.x * TPB + threadIdx.x;
  const long long base = tid * (long long)EPT;
  const float* xg = x   + base;
  const int*   ig = idx + base;

  const unsigned xo0 = lds_off(wbase + lane * 16);
  const unsigned io0 = lds_off(wbase + 512 + lane * 16);

  // Prefetch stage 0 into buffer 0.
  async_cp16_nt(xo0, xg);
  async_cp16_nt(io0, ig);

  int   runSeg = -1;
  float runSum = 0.0f;
  float runCnt = 0.0f;

#pragma unroll
  for (int s = 0; s < STAGES; ++s) {
    const int b = s & 1;
    if (s + 1 < STAGES) {
      // Issue next stage into the other buffer, then wait for current stage.
      async_cp16_nt((b ? xo0 : xo0 + 1024), xg + 4 * (s + 1));
      async_cp16_nt((b ? io0 : io0 + 1024), ig + 4 * (s + 1));
      // Async loads retire in order: <=2 outstanding means the current
      // stage's 16B x + 16B idx per lane have landed in LDS.
      asm volatile("s_wait_asynccnt 2" ::: "memory");
    } else {
      asm volatile("s_wait_asynccnt 0" ::: "memory");
    }

    const float4 xv = *(const float4*)(wbase + b * 1024 + lane * 16);
    const int4   sv = *(const int4*)  (wbase + b * 1024 + 512 + lane * 16);

    const int   ss[4] = {sv.x, sv.y, sv.z, sv.w};
    const float vv[4] = {xv.x, xv.y, xv.z, xv.w};
#pragma unroll
    for (int j = 0; j < 4; ++j) {
      if (ss[j] != runSeg) {
        if (runCnt != 0.0f) {
          atomicAdd(&seg_sum[runSeg], runSum);   // no-return -> STOREcnt path
          atomicAdd(&seg_cnt[runSeg], runCnt);
        }
        runSeg = ss[j];
        runSum = 0.0f;
        runCnt = 0.0f;
      }
      runSum += vv[j];
      runCnt += 1.0f;
    }
  }

  if (runCnt != 0.0f) {
    atomicAdd(&seg_sum[runSeg], runSum);
    atomicAdd(&seg_cnt[runSeg], runCnt);
  }
}

__global__ __launch_bounds__(TPB) void finalize_kernel(const float* __restrict__ ws,
                                                       float* __restrict__ out) {
  int i = blockIdx.x * blockDim.x + threadIdx.x;
  if (i < NSEG) {
    out[i] = ws[i] / fmaxf(ws[NSEG + i], 1.0f);
  }
}

extern "C" void kernel_launch(void* const* d_in, const int* in_sizes, int n_in,
                              void* d_out, int out_size, void* d_ws, size_t ws_size,
                              hipStream_t stream) {
  (void)in_sizes; (void)n_in; (void)out_size; (void)ws_size;

  const float* x   = (const float*)d_in[0];
  const int*   idx = (const int*)d_in[1];
  float*       ws  = (float*)d_ws;              // [0..NSEG) sums, [NSEG..2*NSEG) counts
  float*       out = (float*)d_out;

  zero_ws_kernel<<<(2 * NSEG + TPB - 1) / TPB, TPB, 0, stream>>>(ws, 2 * NSEG);

  const int nthreads = NTOT / EPT;              // 524288, exact
  segsum_kernel<<<nthreads / TPB, TPB, 0, stream>>>(x, idx, ws, ws + NSEG);

  finalize_kernel<<<(NSEG + TPB - 1) / TPB, TPB, 0, stream>>>(ws, out);
}